// DecoderForLarge_75084618269503
// MI455X (gfx1250) — compile-verified
//
#include <hip/hip_runtime.h>
#include <hip/hip_bf16.h>
#include <math.h>

#define B_ 16
#define N_ 10000
#define G_ 100
#define H_ 128
#define GT_ 7          // ceil(G/16) g-tiles
#define KC_ 64         // n-chunk for agg GEMM
#define NCH_ ((N_ + KC_ - 1) / KC_)   // 157 chunks

typedef float v2f __attribute__((ext_vector_type(2)));
typedef float v8f __attribute__((ext_vector_type(8)));

// gfx1250 async copy: global -> LDS, 16B per lane, tracked by ASYNCcnt.
__device__ __forceinline__ void async_g2l_b128(unsigned ldsAddr, int gOffBytes,
                                               const void* sBase) {
    asm volatile("global_load_async_to_lds_b128 %0, %1, %2"
                 :: "v"(ldsAddr), "v"(gOffBytes), "s"(sBase)
                 : "memory");
}

// ---------------------------------------------------------------------------
// small helpers
// ---------------------------------------------------------------------------
__global__ __launch_bounds__(256) void k_zero(float* p, int n) {
    int i = blockIdx.x * 256 + threadIdx.x;
    if (i < n) p[i] = 0.0f;
}

__global__ __launch_bounds__(128) void k_mean_partial(const float* __restrict__ E,
                                                      float* __restrict__ meanE) {
    const int b = blockIdx.x;
    const int slice = blockIdx.y;      // 0..63
    const int h = threadIdx.x;         // 0..127
    float s = 0.0f;
    for (int n = slice; n < N_; n += 64)
        s += E[((size_t)b * N_ + n) * H_ + h];
    atomicAdd(&meanE[b * H_ + h], s);
}

__global__ __launch_bounds__(128) void k_qgraph(const float* __restrict__ meanE,
                                                const float* __restrict__ Wg,
                                                float* __restrict__ qg) {
    const int b = blockIdx.x, o = threadIdx.x;
    __shared__ float sm[H_];
    sm[o] = meanE[b * H_ + o];
    __syncthreads();
    float s = 0.0f;
    for (int h = 0; h < H_; ++h) s += sm[h] * Wg[o * H_ + h];
    qg[b * H_ + o] = s * (1.0f / (float)N_);
}

__global__ __launch_bounds__(128) void k_qinit(const float* __restrict__ E,
                                               const float* __restrict__ coords,
                                               const int* __restrict__ last,
                                               const float* __restrict__ Wl,
                                               const float* __restrict__ Wf,
                                               const float* __restrict__ qg,
                                               float* __restrict__ qsum,
                                               float* __restrict__ lastc) {
    const int g = blockIdx.x, b = blockIdx.y, o = threadIdx.x;
    const int ln = last[b * G_ + g];
    __shared__ float se[H_];
    se[o] = E[((size_t)b * N_ + ln) * H_ + o];
    __syncthreads();
    float s = 0.0f;
    for (int h = 0; h < H_; ++h) s += se[h] * (Wl[o * H_ + h] + Wf[o * H_ + h]);
    qsum[((size_t)b * G_ + g) * H_ + o] = s + qg[b * H_ + o];
    if (o < 2) lastc[((size_t)b * G_ + g) * 2 + o] = coords[((size_t)b * N_ + ln) * 2 + o];
}

__global__ __launch_bounds__(128) void k_final(const float* __restrict__ agg,
                                               const float* __restrict__ Wv,
                                               const float* __restrict__ qsum,
                                               float* __restrict__ finalq) {
    const int g = blockIdx.x, b = blockIdx.y, o = threadIdx.x;
    __shared__ float sa[H_];
    sa[o] = agg[((size_t)b * G_ + g) * H_ + o];
    __syncthreads();
    float s = 0.0f;
    for (int h = 0; h < H_; ++h) s += sa[h] * Wv[o * H_ + h];
    finalq[((size_t)b * G_ + g) * H_ + o] = s + qsum[((size_t)b * G_ + g) * H_ + o];
}

// ---------------------------------------------------------------------------
// big GEMM 1: agg[b,g,h] = (1/N) sum_n mask01[b,g,n] * E[b,n,h]
// Double-buffered: async global->LDS copies for E overlap the WMMA compute;
// one barrier per chunk.
// ---------------------------------------------------------------------------
__global__ __launch_bounds__(256) void k_agg(const float* __restrict__ E,
                                             const float* __restrict__ mask,
                                             float* __restrict__ agg) {
    __shared__ float sE[2][KC_][H_];       // 64 KB double-buffered E chunk
    __shared__ float sM[2][16][KC_ + 4];   // padded stride 68
    const int gt = blockIdx.x, b = blockIdx.y;
    const int tid = threadIdx.x, lane = tid & 31, wave = tid >> 5;
    const int g0 = gt * 16;

    const float* Eb = E + (size_t)b * N_ * H_;
    const float* Mb = mask + (size_t)b * G_ * N_;

    v8f acc = {};
    const int gA  = lane & 15;
    const int kHi = (lane >> 4) * 2;
    const int hB  = (wave << 4) + (lane & 15);

    float mreg[4];

    // issue async copy of one 64xH E chunk into buffer bufIdx (8 x B128 / thread)
    auto issueE = [&](int nb, int bufIdx) {
        unsigned ldsBase = (unsigned)(size_t)(&sE[bufIdx][0][0]);
#pragma unroll
        for (int i = 0; i < 8; ++i) {
            int f = tid * 4 + i * 1024;          // float index within chunk
            int n = f >> 7, col = f & 127;
            int row = nb + n; if (row > N_ - 1) row = N_ - 1;   // clamp tail (masked to 0)
            int gOff = (row * H_ + col) * 4;
            async_g2l_b128(ldsBase + (unsigned)f * 4u, gOff, Eb);
        }
    };
    auto loadMask = [&](int nb) {
#pragma unroll
        for (int j = 0; j < 4; ++j) {
            int idx = tid + j * 256;             // 0..1023 of 16x64 tile
            int g = idx >> 6, n = idx & 63;
            float m = 0.0f;
            int gg = g0 + g, nn = nb + n;
            if (gg < G_ && nn < N_) {
                float raw = Mb[(size_t)gg * N_ + nn];
                m = (raw < -1.0e30f) ? 1.0f : 0.0f;   // -inf => visited => 1
            }
            mreg[j] = m;
        }
    };
    auto storeMask = [&](int bufIdx) {
#pragma unroll
        for (int j = 0; j < 4; ++j) {
            int idx = tid + j * 256;
            int g = idx >> 6, n = idx & 63;
            sM[bufIdx][g][n] = mreg[j];
        }
    };

    // prologue: chunk 0 into buffer 0
    issueE(0, 0);
    loadMask(0);
    storeMask(0);
    asm volatile("s_wait_asynccnt 0x0" ::: "memory");
    __syncthreads();

    int buf = 0;
    for (int c = 0; c < NCH_; ++c) {
        if (c + 1 < NCH_) {                      // prefetch next chunk into other buffer
            issueE((c + 1) * KC_, buf ^ 1);
            loadMask((c + 1) * KC_);
            storeMask(buf ^ 1);
        }
        // ---- 16 f32 WMMA steps on current buffer ----
#pragma unroll
        for (int k4 = 0; k4 < KC_; k4 += 4) {
            v2f a, bm;
            a.x  = sM[buf][gA][k4 + kHi];
            a.y  = sM[buf][gA][k4 + kHi + 1];
            bm.x = sE[buf][k4 + kHi][hB];
            bm.y = sE[buf][k4 + kHi + 1][hB];
            acc = __builtin_amdgcn_wmma_f32_16x16x4_f32(false, a, false, bm,
                                                        (short)0, acc, false, false);
        }
        if (c + 1 < NCH_)
            asm volatile("s_wait_asynccnt 0x0" ::: "memory");
        __syncthreads();
        buf ^= 1;
    }

    const float invN = 1.0f / (float)N_;
    const int gBase = g0 + ((lane >> 4) << 3);
#pragma unroll
    for (int r = 0; r < 8; ++r) {
        int g = gBase + r;
        if (g < G_) agg[((size_t)b * G_ + g) * H_ + hB] = acc[r] * invN;
    }
}

// ---------------------------------------------------------------------------
// big GEMM 2 + epilogue: score = fq @ E^T / sqrt(H) - dist/sqrt(2);
// clipped = 10*tanh; +(-1e8 if masked); softmax with fixed max = 10.0.
// Each wave owns a double-buffered LDS E-tile region -> async pipeline with
// no cross-wave barriers in the main loop.
// ---------------------------------------------------------------------------
__global__ __launch_bounds__(256) void k_score_softmax(const float* __restrict__ E,
                                                       const float* __restrict__ coords,
                                                       const float* __restrict__ mask,
                                                       const float* __restrict__ finalq,
                                                       const float* __restrict__ lastc,
                                                       float* __restrict__ out) {
    __shared__ float sQ[16][H_ + 4];             // padded stride 132
    __shared__ float sEw[8][2][16][H_ + 4];      // per-wave double-buffered E tiles
    __shared__ float sLC[16][2];
    __shared__ float sSum[16];

    const int gt = blockIdx.x, b = blockIdx.y;
    const int tid = threadIdx.x, lane = tid & 31, wave = tid >> 5;
    const int g0 = gt * 16;

    for (int i = tid; i < 16 * H_; i += 256) {
        int g = i / H_, h = i % H_;
        float v = 0.0f;
        if (g0 + g < G_) v = finalq[((size_t)b * G_ + g0 + g) * H_ + h];
        sQ[g][h] = v;
    }
    if (tid < 32) {
        int g = tid >> 1, c = tid & 1;
        float v = 0.0f;
        if (g0 + g < G_) v = lastc[((size_t)b * G_ + g0 + g) * 2 + c];
        sLC[g][c] = v;
    }
    if (tid < 16) sSum[tid] = 0.0f;
    __syncthreads();

    const float* Eb = E + (size_t)b * N_ * H_;
    const float* Cb = coords + (size_t)b * N_ * 2;
    const float* Mb = mask + (size_t)b * G_ * N_;
    float* Ob = out + (size_t)b * G_ * N_;

    const int gA  = lane & 15;
    const int kHi = (lane >> 4) * 2;
    const int nIn = lane & 15;
    const int gBase = g0 + ((lane >> 4) << 3);
    const float invSqrtH = 0.08838834764831844f;   // 1/sqrt(128)
    const float invSqrt2 = 0.7071067811865476f;

    // async copy one 16xH tile into this wave's buffer (16 x B128 / lane)
    auto issueTile = [&](int nb0, int bufIdx) {
        unsigned ldsBase = (unsigned)(size_t)(&sEw[wave][bufIdx][0][0]);
#pragma unroll
        for (int i = 0; i < 16; ++i) {
            int row = nb0 + i; if (row > N_ - 1) row = N_ - 1;   // clamp: cols n>=N discarded
            int gOff = (row * H_ + lane * 4) * 4;
            async_g2l_b128(ldsBase + (unsigned)(i * (H_ + 4) + lane * 4) * 4u, gOff, Eb);
        }
    };

    float psum[8];
#pragma unroll
    for (int r = 0; r < 8; ++r) psum[r] = 0.0f;

    int buf = 0;
    issueTile(wave * 16, 0);

    for (int nb0 = wave * 16; nb0 < N_; nb0 += 8 * 16) {
        const int nb1 = nb0 + 8 * 16;
        const bool more = (nb1 < N_);            // wave-uniform
        if (more) {
            issueTile(nb1, buf ^ 1);
            asm volatile("s_wait_asynccnt 0x10" ::: "memory");  // current tile done
        } else {
            asm volatile("s_wait_asynccnt 0x0" ::: "memory");
        }

        // ---- 32 f32 WMMA steps, K = H = 128 ----
        v8f acc = {};
#pragma unroll
        for (int k4 = 0; k4 < H_; k4 += 4) {
            v2f a, bm;
            a.x  = sQ[gA][k4 + kHi];
            a.y  = sQ[gA][k4 + kHi + 1];
            bm.x = sEw[wave][buf][nIn][k4 + kHi];
            bm.y = sEw[wave][buf][nIn][k4 + kHi + 1];
            acc = __builtin_amdgcn_wmma_f32_16x16x4_f32(false, a, false, bm,
                                                        (short)0, acc, false, false);
        }

        // ---- fused epilogue ----
        const int n = nb0 + nIn;
        const bool nOK = (n < N_);
        float cx = 0.f, cy = 0.f;
        if (nOK) { cx = Cb[n * 2 + 0]; cy = Cb[n * 2 + 1]; }
#pragma unroll
        for (int r = 0; r < 8; ++r) {
            int g = gBase + r;
            if (g < G_ && nOK) {
                int gl = g - g0;
                float dx = sLC[gl][0] - cx, dy = sLC[gl][1] - cy;
                float dist = sqrtf(fmaxf(dx * dx + dy * dy, 1e-12f));
                float s  = acc[r] * invSqrtH - dist * invSqrt2;
                float sc = 10.0f * tanhf(s);
                float raw = Mb[(size_t)g * N_ + n];
                if (raw < -1.0e30f) sc += -1.0e8f;
                float e = expf(sc - 10.0f);        // fixed max: clipped score <= 10
                psum[r] += e;
                Ob[(size_t)g * N_ + n] = e;
            }
        }
        buf ^= 1;
    }

    // ---- block-wide row sums ----
#pragma unroll
    for (int r = 0; r < 8; ++r) {
        int g = gBase + r;
        if (g < G_) atomicAdd(&sSum[g - g0], psum[r]);
    }
    __syncthreads();

    float inv[8];
#pragma unroll
    for (int r = 0; r < 8; ++r) {
        int g = gBase + r;
        inv[r] = (g < G_) ? (1.0f / sSum[g - g0]) : 0.0f;
    }

    // ---- normalize: each thread re-touches exactly its own writes ----
    for (int nb0 = wave * 16; nb0 < N_; nb0 += 8 * 16) {
        const int n = nb0 + nIn;
        if (n >= N_) continue;
#pragma unroll
        for (int r = 0; r < 8; ++r) {
            int g = gBase + r;
            if (g < G_) Ob[(size_t)g * N_ + n] *= inv[r];
        }
    }
}

// ---------------------------------------------------------------------------
extern "C" void kernel_launch(void* const* d_in, const int* in_sizes, int n_in,
                              void* d_out, int out_size, void* d_ws, size_t ws_size,
                              hipStream_t stream) {
    const float* E      = (const float*)d_in[0];  // [B,N,H]
    const float* coords = (const float*)d_in[1];  // [B,N,2]
    const int*   last   = (const int*)  d_in[2];  // [B,G]
    const float* mask   = (const float*)d_in[3];  // [B,G,N]
    const float* Wg     = (const float*)d_in[4];
    const float* Wf     = (const float*)d_in[5];
    const float* Wl     = (const float*)d_in[6];
    const float* Wv     = (const float*)d_in[7];
    float* out = (float*)d_out;

    float* ws     = (float*)d_ws;
    float* meanE  = ws;                       // B*H
    float* qg     = meanE  + B_ * H_;         // B*H
    float* qsum   = qg     + B_ * H_;         // B*G*H
    float* agg    = qsum   + (size_t)B_ * G_ * H_;
    float* finalq = agg    + (size_t)B_ * G_ * H_;
    float* lastc  = finalq + (size_t)B_ * G_ * H_;   // B*G*2

    k_zero<<<dim3((B_ * H_ + 255) / 256), dim3(256), 0, stream>>>(meanE, B_ * H_);
    k_mean_partial<<<dim3(B_, 64), dim3(128), 0, stream>>>(E, meanE);
    k_qgraph<<<dim3(B_), dim3(128), 0, stream>>>(meanE, Wg, qg);
    k_qinit<<<dim3(G_, B_), dim3(128), 0, stream>>>(E, coords, last, Wl, Wf, qg, qsum, lastc);
    k_agg<<<dim3(GT_, B_), dim3(256), 0, stream>>>(E, mask, agg);
    k_final<<<dim3(G_, B_), dim3(128), 0, stream>>>(agg, Wv, qsum, finalq);
    k_score_softmax<<<dim3(GT_, B_), dim3(256), 0, stream>>>(E, coords, mask, finalq, lastc, out);
}